// GCNBaseline_17059610100486
// MI455X (gfx1250) — compile-verified
//
#include <hip/hip_runtime.h>

typedef __attribute__((ext_vector_type(2))) float v2f;
typedef __attribute__((ext_vector_type(8))) float v8f;

#define GCN_NIN 256
#define GCN_HID 64

// ---------------- degree / normalization ----------------

__global__ void k_init_deg(float* __restrict__ deg, int n) {
  int i = blockIdx.x * blockDim.x + threadIdx.x;
  if (i < n) deg[i] = 1.0f;  // self-loop contributes 1
}

__global__ void k_accum_deg(const int* __restrict__ dst, float* __restrict__ deg, int nE) {
  int e = blockIdx.x * blockDim.x + threadIdx.x;
  if (e < nE) atomicAdd(&deg[dst[e]], 1.0f);
}

__global__ void k_dinv(const float* __restrict__ deg, float* __restrict__ dinv, int n) {
  int i = blockIdx.x * blockDim.x + threadIdx.x;
  if (i < n) {
    float d = deg[i];
    dinv[i] = d > 0.0f ? rsqrtf(d) : 0.0f;
  }
}

// ---------------- f32 WMMA GEMM: C[MxNOUT] = A[MxK] @ W[KxNOUT] (+bias)(+relu) -------------
// One wave per 16x16 output tile. W staged in LDS. M must be a multiple of 16.

template <int K, int NOUT, bool BIAS, bool RELU>
__global__ __launch_bounds__(256) void k_gemm_wmma(const float* __restrict__ A,
                                                   const float* __restrict__ W,
                                                   const float* __restrict__ bias,
                                                   float* __restrict__ C, int M) {
  __shared__ float sW[K * NOUT];
  for (int idx = threadIdx.x; idx < K * NOUT; idx += 256) sW[idx] = W[idx];
  __syncthreads();

  const int tilesAcross = NOUT / 16;
  const int totalTiles = (M / 16) * tilesAcross;
  int wave = blockIdx.x * 8 + (threadIdx.x >> 5);
  if (wave >= totalTiles) return;  // wave-uniform: EXEC stays all-ones below

  int tm = wave / tilesAcross;
  int tn = wave % tilesAcross;
  int lane = threadIdx.x & 31;
  int half = lane >> 4;    // 0: lanes 0-15, 1: lanes 16-31
  int l15 = lane & 15;
  int khalf = half * 2;    // A/B VGPRs: K = k0 + vgpr + 2*half
  int row = tm * 16 + l15; // A-matrix M index (same rows in both halves)
  int col = tn * 16 + l15; // B/C/D N index

  const float* arow = A + (size_t)row * K + khalf;
  v8f acc = {};
#pragma unroll 4
  for (int k0 = 0; k0 < K; k0 += 4) {
    v2f a = *(const v2f*)(arow + k0);                 // {K=k0+khalf, K=k0+khalf+1}
    v2f b;
    b.x = sW[(k0 + khalf) * NOUT + col];              // conflict-free: col-contiguous
    b.y = sW[(k0 + khalf + 1) * NOUT + col];
    acc = __builtin_amdgcn_wmma_f32_16x16x4_f32(false, a, false, b, (short)0, acc,
                                                false, false);
  }

  float bi = BIAS ? bias[col] : 0.0f;
#pragma unroll
  for (int r = 0; r < 8; ++r) {  // D layout: VGPR r -> M = r + 8*half
    float v = acc[r] + bi;
    if (RELU) v = fmaxf(v, 0.0f);
    C[(size_t)(tm * 16 + half * 8 + r) * NOUT + col] = v;
  }
}

// ---------------- aggregation ----------------

// self-loop term: out[i,:] = hw[i,:] * dinv[i]^2  (also fully initializes out)
__global__ void k_self_term(const float* __restrict__ hw, const float* __restrict__ dinv,
                            float* __restrict__ out, int n) {
  long long t = (long long)blockIdx.x * blockDim.x + threadIdx.x;
  if (t >= (long long)n * GCN_HID) return;
  int i = (int)(t >> 6);
  float di = dinv[i];
  out[t] = hw[t] * di * di;
}

// edge scatter-add: out[dst,:] += hw[src,:] * dinv[src]*dinv[dst]; 64 threads / edge
__global__ void k_edge_agg(const float* __restrict__ hw, const int* __restrict__ src,
                           const int* __restrict__ dst, const float* __restrict__ dinv,
                           float* __restrict__ out, int nE) {
  long long t = (long long)blockIdx.x * blockDim.x + threadIdx.x;
  int e = (int)(t >> 6);
  if (e >= nE) return;
  int j = (int)(t & 63);
  int s = src[e];
  int d = dst[e];
  if (j == 0 && e + 16 < nE) {
    // warm L2/WGP$ for an upcoming gather row (global_prefetch_b8)
    __builtin_prefetch(hw + (size_t)src[e + 16] * GCN_HID, 0, 0);
  }
  float nrm = dinv[s] * dinv[d];
  atomicAdd(out + (size_t)d * GCN_HID + j, hw[(size_t)s * GCN_HID + j] * nrm);
}

// h = relu(h + b) in place
__global__ void k_bias_relu(float* __restrict__ h, const float* __restrict__ b,
                            long long count) {
  long long t = (long long)blockIdx.x * blockDim.x + threadIdx.x;
  if (t < count) h[t] = fmaxf(h[t] + b[t & (GCN_HID - 1)], 0.0f);
}

// out[i,0:2] = t[i,0:32] @ Wh2[32x2] + bh2
__global__ void k_head2(const float* __restrict__ tin, const float* __restrict__ Wh2,
                        const float* __restrict__ bh2, float* __restrict__ out, int n) {
  int i = blockIdx.x * blockDim.x + threadIdx.x;
  if (i >= n) return;
  float a0 = bh2[0], a1 = bh2[1];
  const float* r = tin + (size_t)i * 32;
#pragma unroll
  for (int k = 0; k < 32; ++k) {
    float v = r[k];
    a0 += v * Wh2[k * 2 + 0];
    a1 += v * Wh2[k * 2 + 1];
  }
  out[(size_t)i * 2 + 0] = a0;
  out[(size_t)i * 2 + 1] = a1;
}

// ---------------- launch ----------------

extern "C" void kernel_launch(void* const* d_in, const int* in_sizes, int n_in,
                              void* d_out, int out_size, void* d_ws, size_t ws_size,
                              hipStream_t stream) {
  const float* x   = (const float*)d_in[0];
  const int*   ei  = (const int*)d_in[1];   // edge_index [2, E] int32
  const float* W1  = (const float*)d_in[2];
  const float* b1  = (const float*)d_in[3];
  const float* W2  = (const float*)d_in[4];
  const float* b2  = (const float*)d_in[5];
  const float* Wh1 = (const float*)d_in[6];
  const float* bh1 = (const float*)d_in[7];
  const float* Wh2 = (const float*)d_in[8];
  const float* bh2 = (const float*)d_in[9];

  const int N = in_sizes[0] / GCN_NIN;
  const int E = in_sizes[1] / 2;
  const int* src = ei;
  const int* dst = ei + E;

  float* deg  = (float*)d_ws;
  float* dinv = deg + N;
  float* bufA = dinv + N;                   // N*HID scratch (hw / t)
  float* bufB = bufA + (size_t)N * GCN_HID; // N*HID scratch (agg / h)

  const int T = 256;
  unsigned gN   = (unsigned)((N + T - 1) / T);
  unsigned gE   = (unsigned)((E + T - 1) / T);
  unsigned gNH  = (unsigned)(((long long)N * GCN_HID + T - 1) / T);
  unsigned gEH  = (unsigned)(((long long)E * GCN_HID + T - 1) / T);
  unsigned gT64 = (unsigned)(((N / 16) * 4 + 7) / 8);  // NOUT=64 tiles, 8 waves/block
  unsigned gT32 = (unsigned)(((N / 16) * 2 + 7) / 8);  // NOUT=32 tiles

  // normalization
  k_init_deg<<<gN, T, 0, stream>>>(deg, N);
  k_accum_deg<<<gE, T, 0, stream>>>(dst, deg, E);
  k_dinv<<<gN, T, 0, stream>>>(deg, dinv, N);

  // layer 1: hw = x @ W1 ; agg = D^-1/2 A D^-1/2 hw ; h1 = relu(agg + b1)
  k_gemm_wmma<GCN_NIN, GCN_HID, false, false>
      <<<gT64, T, 0, stream>>>(x, W1, nullptr, bufA, N);
  k_self_term<<<gNH, T, 0, stream>>>(bufA, dinv, bufB, N);
  k_edge_agg<<<gEH, T, 0, stream>>>(bufA, src, dst, dinv, bufB, E);
  k_bias_relu<<<gNH, T, 0, stream>>>(bufB, b1, (long long)N * GCN_HID);

  // layer 2
  k_gemm_wmma<GCN_HID, GCN_HID, false, false>
      <<<gT64, T, 0, stream>>>(bufB, W2, nullptr, bufA, N);
  k_self_term<<<gNH, T, 0, stream>>>(bufA, dinv, bufB, N);
  k_edge_agg<<<gEH, T, 0, stream>>>(bufA, src, dst, dinv, bufB, E);
  k_bias_relu<<<gNH, T, 0, stream>>>(bufB, b2, (long long)N * GCN_HID);

  // head: t = relu(h2 @ Wh1 + bh1) ; out = t @ Wh2 + bh2
  k_gemm_wmma<GCN_HID, GCN_HID / 2, true, true>
      <<<gT32, T, 0, stream>>>(bufB, Wh1, bh1, bufA, N);
  k_head2<<<gN, T, 0, stream>>>(bufA, Wh2, bh2, (float*)d_out, N);
}